// GATNet_38362647888011
// MI455X (gfx1250) — compile-verified
//
#include <hip/hip_runtime.h>
#include <hip/hip_bf16.h>
#include <math.h>

// ---------------- problem constants (match reference) ----------------
#define N_NODES 50000
#define F_INF   35
#define E_RAW   800000
#define E_TOT   850000      // + self loops
#define NGRAPH  1024
#define H1      10
#define C1      35
#define HC1     350         // H1*C1
#define NP1     352         // padded 350 -> /16
#define C2      128
#define K2P     352         // padded K for gemm2 (350 -> 352)

typedef __attribute__((ext_vector_type(2))) float v2f;
typedef __attribute__((ext_vector_type(8))) float v8f;

// monotone float <-> u32 encoding so atomicMax(u32) == float max.
// key 0 is strictly below every real float's encoding (acts as -inf init).
__device__ __forceinline__ unsigned fenc(float f) {
  unsigned u = __float_as_uint(f);
  return (u & 0x80000000u) ? ~u : (u | 0x80000000u);
}
__device__ __forceinline__ float fdec(unsigned u) {
  unsigned b = (u & 0x80000000u) ? (u & 0x7FFFFFFFu) : ~u;
  return __uint_as_float(b);
}
__device__ __forceinline__ float lrelu(float t) { return t > 0.f ? t : 0.2f * t; }

__device__ __forceinline__ void edge_sd(const int* __restrict__ ei, int e, int& s, int& d) {
  if (e < E_RAW) { s = ei[e]; d = ei[E_RAW + e]; }
  else           { s = e - E_RAW; d = s; }         // self loops
}

// ================= CSR build (counting sort by dst) =================
__global__ void k_deg(const int* __restrict__ ei, int* __restrict__ deg) {
  int e = blockIdx.x * blockDim.x + threadIdx.x;
  if (e >= E_TOT) return;
  int s, d; edge_sd(ei, e, s, d);
  atomicAdd(&deg[d], 1);
}

// single-block chunked exclusive scan of deg[N] -> off[N+1]; cursor = off copy
__global__ __launch_bounds__(1024) void k_scan(const int* __restrict__ deg,
                                               int* __restrict__ off,
                                               int* __restrict__ cursor) {
  __shared__ int sc[1024];
  __shared__ int carry;
  if (threadIdx.x == 0) carry = 0;
  __syncthreads();
  for (int base = 0; base < N_NODES; base += 1024) {
    int i = base + threadIdx.x;
    int v = (i < N_NODES) ? deg[i] : 0;
    sc[threadIdx.x] = v;
    __syncthreads();
    for (int ofs = 1; ofs < 1024; ofs <<= 1) {       // inclusive Hillis-Steele
      int t = (threadIdx.x >= ofs) ? sc[threadIdx.x - ofs] : 0;
      __syncthreads();
      sc[threadIdx.x] += t;
      __syncthreads();
    }
    int incl = sc[threadIdx.x];
    int excl = incl - v + carry;                     // reads old carry
    if (i < N_NODES) { off[i] = excl; cursor[i] = excl; }
    __syncthreads();
    if (threadIdx.x == 1023) carry += incl;          // chunk total
    __syncthreads();
  }
  if (threadIdx.x == 0) off[N_NODES] = carry;        // == E_TOT
}

__global__ void k_scatter(const int* __restrict__ ei, int* __restrict__ cursor,
                          int* __restrict__ csr_src) {
  int e = blockIdx.x * blockDim.x + threadIdx.x;
  if (e >= E_TOT) return;
  int s, d; edge_sd(ei, e, s, d);
  int pos = atomicAdd(&cursor[d], 1);
  csr_src[pos] = s;
}

// ======== GEMM1: h1[N,352] = x[N,35] @ W1[35,350] (WMMA f32 16x16x4) ========
__global__ __launch_bounds__(256) void k_gemm1(const float* __restrict__ x,
                                               const float* __restrict__ W1,
                                               float* __restrict__ h1) {
  __shared__ float sW[36 * 352];                    // whole W1, zero padded
  for (int i = threadIdx.x; i < 36 * 352; i += 256) {
    int k = i / 352, n = i % 352;
    sW[i] = (k < F_INF && n < HC1) ? W1[k * HC1 + n] : 0.f;
  }
  __syncthreads();

  const int CT = NP1 / 16;                          // 22 column tiles
  int wave = threadIdx.x >> 5, lane = threadIdx.x & 31;
  int tile = blockIdx.x * 8 + wave;
  if (tile >= (N_NODES / 16) * CT) return;
  int r = tile / CT, c = tile % CT;
  int m  = lane & 15;
  int kh = (lane >> 4) << 1;                        // K offset for upper half-wave

  v8f acc = {};
  const float* xrow = x + (size_t)(r * 16 + m) * F_INF;
  #pragma unroll
  for (int k0 = 0; k0 < 36; k0 += 4) {
    int ka = k0 + kh;
    v2f a, b;
    a.x = (ka     < F_INF) ? xrow[ka]     : 0.f;
    a.y = (ka + 1 < F_INF) ? xrow[ka + 1] : 0.f;
    b.x = sW[(k0 + kh    ) * 352 + c * 16 + m];
    b.y = sW[(k0 + kh + 1) * 352 + c * 16 + m];
    acc = __builtin_amdgcn_wmma_f32_16x16x4_f32(false, a, false, b, (short)0, acc,
                                                false, false);
  }
  int col = c * 16 + m;
  int rowbase = r * 16 + ((lane >> 4) << 3);        // lanes 16-31 hold M = v+8
  #pragma unroll
  for (int v = 0; v < 8; ++v)
    h1[(size_t)(rowbase + v) * NP1 + col] = acc[v];
}

// ================= layer-1 attention coefficients [N,10] =================
__global__ void k_att1(const float* __restrict__ h1, const float* __restrict__ aw_s,
                       const float* __restrict__ aw_d, float* __restrict__ as1,
                       float* __restrict__ ad1) {
  int t = blockIdx.x * blockDim.x + threadIdx.x;
  if (t >= N_NODES * H1) return;
  int n = t / H1, h = t % H1;
  const float* row = h1 + (size_t)n * NP1 + h * C1;
  float s = 0.f, d = 0.f;
  #pragma unroll 5
  for (int c = 0; c < C1; ++c) {
    float v = row[c];
    s += v * aw_s[h * C1 + c];
    d += v * aw_d[h * C1 + c];
  }
  as1[t] = s; ad1[t] = d;
}

// segment softmax stats per (dst, head): gather form, no atomics
__global__ void k_soft1(const int* __restrict__ off, const int* __restrict__ csr_src,
                        const float* __restrict__ as1, const float* __restrict__ ad1,
                        float* __restrict__ m1f, float* __restrict__ den1) {
  int t = blockIdx.x * blockDim.x + threadIdx.x;
  if (t >= N_NODES * H1) return;
  int d = t / H1, h = t % H1;
  int k0 = off[d], k1 = off[d + 1];                 // deg >= 1 (self loop)
  float ad = ad1[t];
  float mx = -INFINITY;
  for (int k = k0; k < k1; ++k)
    mx = fmaxf(mx, lrelu(as1[csr_src[k] * H1 + h] + ad));
  float den = 0.f;
  for (int k = k0; k < k1; ++k)
    den += expf(lrelu(as1[csr_src[k] * H1 + h] + ad) - mx);
  m1f[t] = mx; den1[t] = den;
}

// gather aggregation + bias + ELU; wave32 covers 32 consecutive features of one
// dst -> the per-edge h1[src] gather is one coalesced 128B request per wave.
__global__ void k_agg1(const int* __restrict__ off, const int* __restrict__ csr_src,
                       const float* __restrict__ as1, const float* __restrict__ ad1,
                       const float* __restrict__ m1f, const float* __restrict__ den1,
                       const float* __restrict__ h1, const float* __restrict__ b1,
                       float* __restrict__ out1) {
  int t = blockIdx.x * blockDim.x + threadIdx.x;
  if (t >= N_NODES * NP1) return;
  int d = t / NP1, j = t % NP1;
  if (j >= HC1) { out1[t] = 0.f; return; }          // keep pads zero for GEMM2
  int h = j / C1;
  int k0 = off[d], k1 = off[d + 1];
  float ad = ad1[d * H1 + h];
  float mx = m1f[d * H1 + h];
  float rden = 1.f / den1[d * H1 + h];
  float acc = 0.f;
  for (int k = k0; k < k1; ++k) {
    int s = csr_src[k];
    float e = lrelu(as1[s * H1 + h] + ad);
    acc += expf(e - mx) * h1[(size_t)s * NP1 + j];
  }
  acc = acc * rden + b1[j];
  out1[t] = acc > 0.f ? acc : expm1f(acc);          // ELU fused
}

// ======== GEMM2: h2[N,128] = out1[N,350] @ W2[350,128] (WMMA) ========
__global__ __launch_bounds__(256) void k_gemm2(const float* __restrict__ A,
                                               const float* __restrict__ W2,
                                               float* __restrict__ h2) {
  __shared__ float sB[32 * 128];
  __shared__ float sA[16 * 33];
  int r = blockIdx.x;                                // row tile 0..3124
  int wave = threadIdx.x >> 5, lane = threadIdx.x & 31;
  int m = lane & 15, kh = (lane >> 4) << 1;
  v8f acc = {};

  for (int kp = 0; kp < K2P; kp += 32) {
    __syncthreads();
    for (int i = threadIdx.x; i < 32 * 128; i += 256) {
      int kk = i >> 7, n = i & 127;
      int k = kp + kk;
      sB[i] = (k < HC1) ? W2[k * C2 + n] : 0.f;
    }
    for (int i = threadIdx.x; i < 16 * 32; i += 256) {
      int row = i >> 5, kk = i & 31;
      sA[row * 33 + kk] = A[(size_t)(r * 16 + row) * K2P + kp + kk]; // pads zero
    }
    if (kp + 32 < K2P) {                              // prefetch next W2 panel
      int k = kp + 32 + (threadIdx.x >> 3);
      if (k < HC1) __builtin_prefetch(&W2[k * C2 + (threadIdx.x & 7) * 16], 0, 1);
    }
    __syncthreads();
    #pragma unroll
    for (int ks = 0; ks < 32; ks += 4) {
      v2f a, b;
      a.x = sA[m * 33 + ks + kh];
      a.y = sA[m * 33 + ks + kh + 1];
      b.x = sB[(ks + kh    ) * 128 + wave * 16 + m];
      b.y = sB[(ks + kh + 1) * 128 + wave * 16 + m];
      acc = __builtin_amdgcn_wmma_f32_16x16x4_f32(false, a, false, b, (short)0, acc,
                                                  false, false);
    }
  }
  int col = wave * 16 + m;
  int rowbase = r * 16 + ((lane >> 4) << 3);
  #pragma unroll
  for (int v = 0; v < 8; ++v)
    h2[(size_t)(rowbase + v) * C2 + col] = acc[v];
}

// ================= layer 2 (H=1, C=128) =================
__global__ void k_att2(const float* __restrict__ h2, const float* __restrict__ aw_s,
                       const float* __restrict__ aw_d, float* __restrict__ as2,
                       float* __restrict__ ad2) {
  int n = blockIdx.x * blockDim.x + threadIdx.x;
  if (n >= N_NODES) return;
  const float* row = h2 + (size_t)n * C2;
  float s = 0.f, d = 0.f;
  #pragma unroll 8
  for (int c = 0; c < C2; ++c) { float v = row[c]; s += v * aw_s[c]; d += v * aw_d[c]; }
  as2[n] = s; ad2[n] = d;
}

__global__ void k_soft2(const int* __restrict__ off, const int* __restrict__ csr_src,
                        const float* __restrict__ as2, const float* __restrict__ ad2,
                        float* __restrict__ m2f, float* __restrict__ den2) {
  int d = blockIdx.x * blockDim.x + threadIdx.x;
  if (d >= N_NODES) return;
  int k0 = off[d], k1 = off[d + 1];
  float ad = ad2[d];
  float mx = -INFINITY;
  for (int k = k0; k < k1; ++k) mx = fmaxf(mx, lrelu(as2[csr_src[k]] + ad));
  float den = 0.f;
  for (int k = k0; k < k1; ++k) den += expf(lrelu(as2[csr_src[k]] + ad) - mx);
  m2f[d] = mx; den2[d] = den;
}

// gather aggregation + bias + relu + global max pool (fused, no out2 buffer)
__global__ void k_agg2_pool(const int* __restrict__ off, const int* __restrict__ csr_src,
                            const float* __restrict__ as2, const float* __restrict__ ad2,
                            const float* __restrict__ m2f, const float* __restrict__ den2,
                            const float* __restrict__ h2, const float* __restrict__ b2,
                            const int* __restrict__ batch, unsigned* __restrict__ poolk) {
  int t = blockIdx.x * blockDim.x + threadIdx.x;
  if (t >= N_NODES * C2) return;
  int d = t >> 7, c = t & 127;
  int k0 = off[d], k1 = off[d + 1];
  float ad = ad2[d], mx = m2f[d], rden = 1.f / den2[d];
  float acc = 0.f;
  for (int k = k0; k < k1; ++k) {
    int s = csr_src[k];
    float e = lrelu(as2[s] + ad);
    acc += expf(e - mx) * h2[(size_t)s * C2 + c];     // coalesced per wave
  }
  float v = acc * rden + b2[c];
  v = v > 0.f ? v : 0.f;
  atomicMax(&poolk[(size_t)batch[d] * C2 + c], fenc(v)); // order-independent
}

// ============ dense head: out[g] = relu(pool@Wg+bg)@Wo + bo ============
__global__ __launch_bounds__(128) void k_dense(const unsigned* __restrict__ poolk,
                                               const float* __restrict__ Wg,
                                               const float* __restrict__ bg,
                                               const float* __restrict__ Wo,
                                               const float* __restrict__ bo,
                                               float* __restrict__ out) {
  __shared__ float sp[128];
  __shared__ float sr[128];
  int g = blockIdx.x, c = threadIdx.x;
  unsigned key = poolk[(size_t)g * C2 + c];
  sp[c] = (key == 0u) ? 0.f : fdec(key);             // key==0 <=> empty graph -> 0
  __syncthreads();
  float z = bg[c];
  #pragma unroll 8
  for (int k = 0; k < C2; ++k) z += sp[k] * Wg[k * C2 + c];
  z = z > 0.f ? z : 0.f;
  sr[c] = z * Wo[c];
  __syncthreads();
  for (int sft = 64; sft > 0; sft >>= 1) {
    if (c < sft) sr[c] += sr[c + sft];
    __syncthreads();
  }
  if (c == 0) out[g] = sr[0] + bo[0];
}

// ---------------- host side ----------------
extern "C" void kernel_launch(void* const* d_in, const int* in_sizes, int n_in,
                              void* d_out, int out_size, void* d_ws, size_t ws_size,
                              hipStream_t stream) {
  const float* x   = (const float*)d_in[0];
  const int*   ei  = (const int*)  d_in[1];
  const int*   bat = (const int*)  d_in[2];
  const float* W1  = (const float*)d_in[3];
  const float* aS1 = (const float*)d_in[4];
  const float* aD1 = (const float*)d_in[5];
  const float* b1  = (const float*)d_in[6];
  const float* W2  = (const float*)d_in[7];
  const float* aS2 = (const float*)d_in[8];
  const float* aD2 = (const float*)d_in[9];
  const float* b2  = (const float*)d_in[10];
  const float* Wg  = (const float*)d_in[11];
  const float* bg  = (const float*)d_in[12];
  const float* Wo  = (const float*)d_in[13];
  const float* bo  = (const float*)d_in[14];
  float* out = (float*)d_out;

  // workspace layout (aliased across phases): ~154 MB
  float*    wsf   = (float*)d_ws;
  float*    h12   = wsf;                                  // h1 [N,352], later h2 [N,128]
  float*    out1  = h12 + (size_t)N_NODES * NP1;          // out1 [N,352] (GEMM2 A)
  float*    as1   = out1 + (size_t)N_NODES * NP1;         // [N,10]
  float*    ad1   = as1 + (size_t)N_NODES * H1;           // [N,10]
  float*    m1f   = ad1 + (size_t)N_NODES * H1;           // [N,10]; first N reused as m2f
  float*    den1  = m1f + (size_t)N_NODES * H1;           // [N,10]; first N reused as den2
  float*    as2   = den1 + (size_t)N_NODES * H1;          // [N]
  float*    ad2   = as2 + N_NODES;                        // [N]
  int*      deg   = (int*)(ad2 + N_NODES);                // [N]
  int*      cur   = deg + N_NODES;                        // [N]
  int*      off   = cur + N_NODES;                        // [N+1]
  int*      csr   = off + N_NODES + 1;                    // [E_TOT] src per slot
  unsigned* poolk = (unsigned*)(csr + E_TOT);             // [G,128]

  auto cdiv = [](long long a, long long b) { return (int)((a + b - 1) / b); };

  hipMemsetAsync(deg, 0, (size_t)N_NODES * sizeof(int), stream);
  hipMemsetAsync(poolk, 0, (size_t)NGRAPH * C2 * sizeof(unsigned), stream);

  // ---- CSR build (dst-keyed counting sort), shared by both layers ----
  k_deg<<<cdiv(E_TOT, 256), 256, 0, stream>>>(ei, deg);
  k_scan<<<1, 1024, 0, stream>>>(deg, off, cur);
  k_scatter<<<cdiv(E_TOT, 256), 256, 0, stream>>>(ei, cur, csr);

  // ---- layer 1 ----
  k_gemm1<<<cdiv((N_NODES / 16) * (NP1 / 16), 8), 256, 0, stream>>>(x, W1, h12);
  k_att1<<<cdiv((long long)N_NODES * H1, 256), 256, 0, stream>>>(h12, aS1, aD1, as1, ad1);
  k_soft1<<<cdiv((long long)N_NODES * H1, 256), 256, 0, stream>>>(off, csr, as1, ad1,
                                                                  m1f, den1);
  k_agg1<<<cdiv((long long)N_NODES * NP1, 256), 256, 0, stream>>>(off, csr, as1, ad1,
                                                                  m1f, den1, h12, b1, out1);

  // ---- layer 2 ----
  k_gemm2<<<N_NODES / 16, 256, 0, stream>>>(out1, W2, h12);      // h2 overwrites h1
  k_att2<<<cdiv(N_NODES, 256), 256, 0, stream>>>(h12, aS2, aD2, as2, ad2);
  k_soft2<<<cdiv(N_NODES, 256), 256, 0, stream>>>(off, csr, as2, ad2, m1f, den1);
  k_agg2_pool<<<cdiv((long long)N_NODES * C2, 256), 256, 0, stream>>>(
      off, csr, as2, ad2, m1f, den1, h12, b2, bat, poolk);

  // ---- head ----
  k_dense<<<NGRAPH, 128, 0, stream>>>(poolk, Wg, bg, Wo, bo, out);
}